// NASRNN_24601572671819
// MI455X (gfx1250) — compile-verified
//
#include <hip/hip_runtime.h>
#include <math.h>

typedef __attribute__((ext_vector_type(2))) float v2f;
typedef __attribute__((ext_vector_type(8))) float v8f;
typedef __attribute__((address_space(3))) float lds_float;

#define HIDDEN 512
#define MAT_ELEMS (512 * 512)   // one [H,H] weight matrix
#define HCHUNKS 4               // h-dimension split for memory parallelism
#define HCHUNK_LEN (HIDDEN / HCHUNKS)   // 128 rows per chunk

// ---------------------------------------------------------------------------
// Partial GEMV via V_WMMA_F32_16X16X4_F32 (raw partial sums, no relu).
//   out[k] = sum_{h in chunk} x[h] * W[h][k]
// One 256-thread block (8 waves) covers 256 columns ('half' picks which 256)
// over one 128-row h-chunk. The x-chunk is staged once into LDS with the
// gfx1250 async-to-LDS path, so every global load in the hot loop is weight
// traffic. A-matrix (16x4 f32): x replicated into all 16 M rows (v0 = K0|K2,
// v1 = K1|K3 across half-waves), so every D row holds the GEMV result;
// lanes 0-15 of D's first VGPR give out[k0..k0+15]. B-matrix (4x16 f32):
// v0 = rows K0|K2, v1 = rows K1|K3, N striped over lanes.
// EXEC is all-ones through every WMMA; lane predication only on the store.
// ---------------------------------------------------------------------------
__device__ __forceinline__ void gemv_wmma_partial(const float* __restrict__ x,
                                                  const float* __restrict__ W,
                                                  float* __restrict__ out,
                                                  int half, int hchunk) {
    const int tid    = threadIdx.x;
    const int wave   = tid >> 5;
    const int lane   = tid & 31;
    const int lanelo = lane & 15;
    const int hi     = lane >> 4;          // 0 = low half-wave, 1 = high

    const int kbase = half * 256 + wave * 32;
    const int k0 = kbase;
    const int k1 = kbase + 16;

    const int hbeg = hchunk * HCHUNK_LEN;

    // Stage the 128-float x-chunk into LDS via async DMA (wave 0 only:
    // 32 lanes x b128 = 512 B), then barrier so all waves can read it.
    __shared__ float sx[HCHUNK_LEN];
    if (wave == 0) {
        lds_float*   dst = (lds_float*)&sx[lane * 4];
        const float* src = x + hbeg + lane * 4;
        asm volatile("global_load_async_to_lds_b128 %0, %1, off"
                     :: "v"(dst), "v"(src)
                     : "memory");
        asm volatile("s_wait_asynccnt 0x0" ::: "memory");
    }
    __syncthreads();

    v8f acc0 = {};
    v8f acc1 = {};

    for (int h0 = 0; h0 < HCHUNK_LEN; h0 += 4) {
        const int lr0 = h0 + 2 * hi;       // LDS row feeding v0 (K = 0 or 2)
        const int r0  = hbeg + lr0;        // global row index for weights
        const int r1  = r0 + 1;            // row feeding v1 (K = 1 or 3)

        v2f A;                              // contiguous pair -> ds_load_b64
        A.x = sx[lr0];
        A.y = sx[lr0 + 1];

        v2f B0, B1;
        B0.x = W[(size_t)r0 * HIDDEN + k0 + lanelo];
        B0.y = W[(size_t)r1 * HIDDEN + k0 + lanelo];
        B1.x = W[(size_t)r0 * HIDDEN + k1 + lanelo];
        B1.y = W[(size_t)r1 * HIDDEN + k1 + lanelo];

        acc0 = __builtin_amdgcn_wmma_f32_16x16x4_f32(false, A, false, B0,
                                                     (short)0, acc0, false, false);
        acc1 = __builtin_amdgcn_wmma_f32_16x16x4_f32(false, A, false, B1,
                                                     (short)0, acc1, false, false);
    }

    if (lane < 16) {
        out[k0 + lanelo] = acc0[0];        // D row 0, col lanelo (raw partial)
        out[k1 + lanelo] = acc1[0];
    }
}

// ---------------------------------------------------------------------------
// Phase A: all 5 cells' x@W1 partial GEMVs in one launch (fully parallel).
// grid = 320 blocks: block -> (cell, node, half, hchunk).
// partialA layout: [cell*8+node][hchunk][512].
// ---------------------------------------------------------------------------
__global__ void __launch_bounds__(256)
nas_phaseA(const float* __restrict__ a, const float* __restrict__ b,
           const float* __restrict__ c, const float* __restrict__ d,
           const float* __restrict__ e,
           const float* __restrict__ W1, float* __restrict__ partialA) {
    const int wg     = blockIdx.x;         // 0..319
    const int cell   = wg >> 6;            // 64 blocks per cell
    const int sub    = wg & 63;
    const int node   = sub >> 3;           // 8 blocks per matrix
    const int half   = (sub >> 2) & 1;
    const int hchunk = sub & 3;

    // Uniform scalar select -> stays in SGPRs, loads stay GLOBAL (no FLAT).
    const float* x = (cell == 0) ? a
                   : (cell == 1) ? b
                   : (cell == 2) ? c
                   : (cell == 3) ? d : e;

    const size_t mat = (size_t)cell * 8 + node;
    gemv_wmma_partial(x, W1 + mat * MAT_ELEMS,
                      partialA + (mat * HCHUNKS + hchunk) * HIDDEN,
                      half, hchunk);
}

// ---------------------------------------------------------------------------
// Phase B (per cell i): raw partials of h @ W2[i][node].
// grid = 64 blocks: block -> (node, half, hchunk).
// partialB layout: [node][hchunk][512].
// ---------------------------------------------------------------------------
__global__ void __launch_bounds__(256)
nas_phaseB(const float* __restrict__ h,
           const float* __restrict__ W2cell,
           float* __restrict__ partialB) {
    const int wg     = blockIdx.x;         // 0..63
    const int node   = wg >> 3;
    const int half   = (wg >> 2) & 1;
    const int hchunk = wg & 3;
    gemv_wmma_partial(h, W2cell + (size_t)node * MAT_ELEMS,
                      partialB + ((size_t)node * HCHUNKS + hchunk) * HIDDEN,
                      half, hchunk);
}

// ---------------------------------------------------------------------------
// Combine: fixed-order (deterministic) reduction of BOTH partial sets,
// relu each branch, add, then the NAS activation tree. 1 block x 512 threads.
//   t[n] = relu(sum_c pA[n][c]) + relu(sum_c pB[n][c])
// ---------------------------------------------------------------------------
__device__ __forceinline__ float sigmoidf_(float v) { return 1.0f / (1.0f + expf(-v)); }
__device__ __forceinline__ float reluf_(float v)    { return v > 0.0f ? v : 0.0f; }

__global__ void __launch_bounds__(HIDDEN)
nas_combine(const float* __restrict__ partialB,
            const float* __restrict__ partialAcell,
            float* __restrict__ hout,
            float* __restrict__ finalout) {
    const int k = threadIdx.x;

    float t[8];
#pragma unroll
    for (int n = 0; n < 8; ++n) {
        const float* pB = partialB     + (size_t)n * HCHUNKS * HIDDEN;
        const float* pA = partialAcell + (size_t)n * HCHUNKS * HIDDEN;
        float sB = pB[0 * HIDDEN + k];
        sB += pB[1 * HIDDEN + k];
        sB += pB[2 * HIDDEN + k];
        sB += pB[3 * HIDDEN + k];
        float sA = pA[0 * HIDDEN + k];
        sA += pA[1 * HIDDEN + k];
        sA += pA[2 * HIDDEN + k];
        sA += pA[3 * HIDDEN + k];
        t[n] = reluf_(sA) + reluf_(sB);    // relu(x@W1) + relu(h@W2)
    }

    const float m0 = reluf_(t[0]) + sigmoidf_(t[3]);
    const float m1 = sigmoidf_(t[1]) + tanhf(t[2]);
    const float m2 = sigmoidf_(t[4]) * tanhf(t[5]);
    const float m3 = sigmoidf_(t[6]) * reluf_(t[7]);
    const float m4 = sigmoidf_(m1) + tanhf(m2);
    const float m5 = tanhf(m0) * tanhf(m3);
    const float m6 = tanhf(m4) * tanhf(m5);
    const float r  = tanhf(m6);

    hout[k] = r;
    if (finalout) finalout[k] = r;
}

// ---------------------------------------------------------------------------
extern "C" void kernel_launch(void* const* d_in, const int* in_sizes, int n_in,
                              void* d_out, int out_size, void* d_ws, size_t ws_size,
                              hipStream_t stream) {
    const float* a     = (const float*)d_in[0];
    const float* b     = (const float*)d_in[1];
    const float* c     = (const float*)d_in[2];
    const float* d     = (const float*)d_in[3];
    const float* e     = (const float*)d_in[4];
    const float* state = (const float*)d_in[5];
    const float* W1    = (const float*)d_in[6];
    const float* W2    = (const float*)d_in[7];
    float* out = (float*)d_out;

    float* ws       = (float*)d_ws;
    float* partialA = ws;                                   // 40*4*512 = 81920
    float* partialB = partialA + 40 * HCHUNKS * HIDDEN;     //  8*4*512 = 16384
    float* hbuf     = partialB + 8 * HCHUNKS * HIDDEN;      //  512

    // Parallel precompute of every x_i @ W1[i][n] (40 MiB streamed, 320 WGs).
    nas_phaseA<<<5 * 8 * 2 * HCHUNKS, 256, 0, stream>>>(a, b, c, d, e, W1, partialA);

    // Sequential recurrence: 5 x (GEMV over W2[i] + reduce/combine).
    const float* hin = state;
    for (int i = 0; i < 5; ++i) {
        nas_phaseB<<<8 * 2 * HCHUNKS, 256, 0, stream>>>(hin,
                                                        W2 + (size_t)i * 8 * MAT_ELEMS,
                                                        partialB);
        nas_combine<<<1, HIDDEN, 0, stream>>>(partialB,
                                              partialA + (size_t)i * 8 * HCHUNKS * HIDDEN,
                                              hbuf, (i == 4) ? out : nullptr);
        hin = hbuf;
    }
}